// SpliceGraph_25572235281101
// MI455X (gfx1250) — compile-verified
//
#include <hip/hip_runtime.h>
#include <math.h>

#define N_NODES 10000
#define LIN     4995
#define L2      1247
#define L3      207
#define RFEAT   207
#define KPAD    208
#define HDIM    128
#define NEDGE   320000
#define EPS     1e-5f

typedef float v2f __attribute__((ext_vector_type(2)));
typedef float v8f __attribute__((ext_vector_type(8)));

// ---------------- workspace layout (float offsets) ----------------
#define P_W2EFF   0      // 176 : folded conv1+bn1+conv2 weights [4][4][11]
#define P_B2EFF   176    // 4
#define P_A2      180    // 4   : bn2 scale
#define P_D2      184    // 4   : bn2 shift
#define P_A3      188    // 1   : bn3 scale
#define P_D3      189    // 1   : bn3 shift
#define S_BASE    256
#define S_S1      256    // 4   : sum x per channel
#define S_S2      260    // 10  : packed upper-tri sum x_ci*x_cj
#define S_SUM2    276    // 4   : conv2 out sums
#define S_SSQ2    280    // 4   : conv2 out sumsq
#define S_S3      284    // 1
#define S_SS3     285    // 1
#define S_BN4SUM  288    // 128
#define S_BN4SSQ  416    // 128
#define S_END     544
#define WS_H2     1024
#define WS_FEAT   (WS_H2   + N_NODES*4*L2)     // N*KPAD (feat, bn3-applied, padded)
#define WS_XW     (WS_FEAT + N_NODES*KPAD)     // N*128
#define WS_XL     (WS_XW   + N_NODES*HDIM)
#define WS_Z      (WS_XL   + N_NODES*HDIM)
#define WS_AGG    (WS_Z    + N_NODES*HDIM)
#define WS_DIS    (WS_AGG  + N_NODES*HDIM)     // N
#define WS_WPAD   (WS_DIS  + N_NODES)          // 2 * KPAD*HDIM (padded gcn_w, lin1_w)

// ---------------- block reduction helper ----------------
template<int NV>
__device__ __forceinline__ void block_reduce_add(float (&v)[NV], float* dst) {
  __shared__ float red[8][NV];
  int lane = threadIdx.x & 31;
  int warp = threadIdx.x >> 5;
#pragma unroll
  for (int i = 0; i < NV; ++i) {
#pragma unroll
    for (int off = 16; off > 0; off >>= 1)
      v[i] += __shfl_down(v[i], off, 32);
  }
  if (lane == 0) {
#pragma unroll
    for (int i = 0; i < NV; ++i) red[warp][i] = v[i];
  }
  __syncthreads();
  int nw = blockDim.x >> 5;
  if ((int)threadIdx.x < NV) {
    float a = 0.f;
    for (int w = 0; w < nw; ++w) a += red[w][threadIdx.x];
    atomicAdd(&dst[threadIdx.x], a);
  }
}

// ---------------- kernels ----------------
__global__ void k_zero(float* ws) {
  int idx = blockIdx.x * blockDim.x + threadIdx.x;
  if (idx < (S_END - S_BASE)) ws[S_BASE + idx] = 0.f;
}

// Pass over x: per-channel sums + 4x4 channel second moments (packed UT).
__global__ void k_xstats(const float* __restrict__ x, float* __restrict__ ws) {
  float v[14];
#pragma unroll
  for (int i = 0; i < 14; ++i) v[i] = 0.f;
  int total = N_NODES * LIN;
  for (int t = blockIdx.x * blockDim.x + threadIdx.x; t < total;
       t += gridDim.x * blockDim.x) {
    int n = t / LIN, l = t - n * LIN;
    const float* xb = x + n * 4 * LIN + l;
    float c0 = xb[0], c1 = xb[LIN], c2 = xb[2 * LIN], c3 = xb[3 * LIN];
    v[0] += c0; v[1] += c1; v[2] += c2; v[3] += c3;
    v[4] += c0 * c0; v[5] += c0 * c1; v[6] += c0 * c2; v[7] += c0 * c3;
    v[8] += c1 * c1; v[9] += c1 * c2; v[10] += c1 * c3;
    v[11] += c2 * c2; v[12] += c2 * c3; v[13] += c3 * c3;
  }
  block_reduce_add<14>(v, ws + S_S1);  // S_S1..S_S1+13 contiguous
}

// Derive bn1 analytically; fold conv1+bn1 into conv2 -> W2eff / bias2eff.
__global__ void k_prep1(const float* __restrict__ w1, const float* __restrict__ b1c,
                        const float* __restrict__ w2, const float* __restrict__ b2c,
                        const float* __restrict__ g1, const float* __restrict__ bb1,
                        float* __restrict__ ws) {
  if (threadIdx.x != 0 || blockIdx.x != 0) return;
  const float M = (float)N_NODES * (float)LIN;
  float mx[4];
  for (int c = 0; c < 4; ++c) mx[c] = ws[S_S1 + c] / M;
  float cov[4][4];
  int t = 0;
  for (int ci = 0; ci < 4; ++ci)
    for (int cj = ci; cj < 4; ++cj) {
      float cc = ws[S_S2 + t] / M - mx[ci] * mx[cj];
      cov[ci][cj] = cc; cov[cj][ci] = cc; ++t;
    }
  float A1[4][4], B1[4];
  for (int c = 0; c < 4; ++c) {
    float mean1 = b1c[c];
    for (int ci = 0; ci < 4; ++ci) mean1 += w1[c * 4 + ci] * mx[ci];
    float var1 = 0.f;
    for (int ci = 0; ci < 4; ++ci)
      for (int cj = 0; cj < 4; ++cj)
        var1 += w1[c * 4 + ci] * w1[c * 4 + cj] * cov[ci][cj];
    float a1 = g1[c] * rsqrtf(var1 + EPS);
    for (int ci = 0; ci < 4; ++ci) A1[c][ci] = a1 * w1[c * 4 + ci];
    B1[c] = a1 * (b1c[c] - mean1) + bb1[c];
  }
  for (int co = 0; co < 4; ++co) {
    for (int cx = 0; cx < 4; ++cx)
      for (int j = 0; j < 11; ++j) {
        float acc = 0.f;
        for (int ci = 0; ci < 4; ++ci)
          acc += w2[(co * 4 + ci) * 11 + j] * A1[ci][cx];
        ws[P_W2EFF + (co * 4 + cx) * 11 + j] = acc;
      }
    float acc = b2c[co];
    for (int ci = 0; ci < 4; ++ci)
      for (int j = 0; j < 11; ++j) acc += w2[(co * 4 + ci) * 11 + j] * B1[ci];
    ws[P_B2EFF + co] = acc;
  }
}

// conv2 over folded weights, reading x directly; emits bn2 partial stats.
__global__ void k_conv2(const float* __restrict__ x, float* __restrict__ ws) {
  __shared__ float wsm[180];
  int tid = threadIdx.x;
  if (tid < 180) wsm[tid] = ws[P_W2EFF + tid];
  __syncthreads();
  int idx = blockIdx.x * blockDim.x + tid;
  float sv[8];
#pragma unroll
  for (int i = 0; i < 8; ++i) sv[i] = 0.f;
  if (idx < N_NODES * L2) {
    int n = idx / L2, p = idx - n * L2;
    const float* xb = x + n * 4 * LIN + 4 * p;
    float acc[4];
#pragma unroll
    for (int c = 0; c < 4; ++c) acc[c] = wsm[176 + c];
#pragma unroll
    for (int cx = 0; cx < 4; ++cx) {
      const float* xc = xb + cx * LIN;
#pragma unroll
      for (int j = 0; j < 11; ++j) {
        float xv = xc[j];
#pragma unroll
        for (int c = 0; c < 4; ++c) acc[c] += wsm[(c * 4 + cx) * 11 + j] * xv;
      }
    }
    float* h2 = ws + WS_H2 + n * 4 * L2 + p;
#pragma unroll
    for (int c = 0; c < 4; ++c) {
      h2[c * L2] = acc[c];
      sv[c] = acc[c];
      sv[4 + c] = acc[c] * acc[c];
    }
  }
  block_reduce_add<8>(sv, ws + S_SUM2);  // sums then sumsqs, contiguous
}

__global__ void k_prep2(const float* __restrict__ g2, const float* __restrict__ bb2,
                        float* __restrict__ ws) {
  int c = threadIdx.x;
  if (c >= 4 || blockIdx.x != 0) return;
  const float M = (float)N_NODES * (float)L2;
  float mean = ws[S_SUM2 + c] / M;
  float var = ws[S_SSQ2 + c] / M - mean * mean;
  float a = g2[c] * rsqrtf(var + EPS);
  ws[P_A2 + c] = a;
  ws[P_D2 + c] = bb2[c] - mean * a;
}

// bn2 + relu + conv3 (stride 6, K=11, 4->1 ch); emits bn3 partial stats.
// Writes raw (pre-bn3) feat with row stride KPAD.
__global__ void k_conv3(const float* __restrict__ w3, const float* __restrict__ b3c,
                        float* __restrict__ ws) {
  __shared__ float wl[52];
  int tid = threadIdx.x;
  if (tid < 44) wl[tid] = w3[tid];
  else if (tid < 48) wl[tid] = ws[P_A2 + tid - 44];
  else if (tid < 52) wl[tid] = ws[P_D2 + tid - 48];
  __syncthreads();
  int idx = blockIdx.x * blockDim.x + tid;
  float sv[2] = {0.f, 0.f};
  if (idx < N_NODES * L3) {
    int n = idx / L3, q = idx - n * L3;
    const float* h2 = ws + WS_H2 + n * 4 * L2 + 6 * q;
    float acc = b3c[0];
#pragma unroll
    for (int ci = 0; ci < 4; ++ci) {
      float a = wl[44 + ci], d = wl[48 + ci];
      const float* hc = h2 + ci * L2;
#pragma unroll
      for (int j = 0; j < 11; ++j) {
        float r = fmaxf(0.f, a * hc[j] + d);
        acc += wl[ci * 11 + j] * r;
      }
    }
    ws[WS_FEAT + n * KPAD + q] = acc;
    sv[0] = acc;
    sv[1] = acc * acc;
  }
  block_reduce_add<2>(sv, ws + S_S3);
}

__global__ void k_prep3(const float* __restrict__ g3, const float* __restrict__ bb3,
                        float* __restrict__ ws) {
  if (threadIdx.x != 0 || blockIdx.x != 0) return;
  const float M = (float)N_NODES * (float)L3;
  float mean = ws[S_S3] / M;
  float var = ws[S_SS3] / M - mean * mean;
  float a = g3[0] * rsqrtf(var + EPS);
  ws[P_A3] = a;
  ws[P_D3] = bb3[0] - mean * a;
}

// Apply bn3 affine to feat in place; zero the pad column (col 207).
__global__ void k_featbn(float* __restrict__ ws) {
  int idx = blockIdx.x * blockDim.x + threadIdx.x;
  if (idx < N_NODES * KPAD) {
    int col = idx % KPAD;
    float a3 = ws[P_A3], d3 = ws[P_D3];
    float v = (col < RFEAT) ? fmaf(a3, ws[WS_FEAT + idx], d3) : 0.f;
    ws[WS_FEAT + idx] = v;
  }
}

// Copy gcn_w / lin1_w into zero-padded [KPAD][HDIM] workspace matrices.
__global__ void k_padw(const float* __restrict__ gcn_w,
                       const float* __restrict__ lin1_w, float* __restrict__ ws) {
  int idx = blockIdx.x * blockDim.x + threadIdx.x;
  if (idx < 2 * KPAD * HDIM) {
    int half = idx / (KPAD * HDIM);
    int r = idx - half * (KPAD * HDIM);
    const float* src = half ? lin1_w : gcn_w;
    ws[WS_WPAD + idx] = (r < RFEAT * HDIM) ? src[r] : 0.f;
  }
}

__global__ void k_deginit(float* __restrict__ ws) {
  int n = blockIdx.x * blockDim.x + threadIdx.x;
  if (n < N_NODES) ws[WS_DIS + n] = 1.0f;  // self loop
}

__global__ void k_degacc(const int* __restrict__ ei, float* __restrict__ ws) {
  int e = blockIdx.x * blockDim.x + threadIdx.x;
  if (e < NEDGE) atomicAdd(&ws[WS_DIS + ei[NEDGE + e]], 1.0f);
}

__global__ void k_degfin(float* __restrict__ ws) {
  int n = blockIdx.x * blockDim.x + threadIdx.x;
  if (n < N_NODES) ws[WS_DIS + n] = rsqrtf(ws[WS_DIS + n]);
}

// WMMA GEMM: featbn [N,KPAD] x Wpad [KPAD,128]. Branch-free K loop.
// grid.z: 0 -> xw (gcn, no bias), 1 -> xl (lin1 + bias). One wave per tile.
__global__ void k_gemm_feat(const float* __restrict__ lin1_b,
                            float* __restrict__ ws) {
  int lane = threadIdx.x;
  int mt = blockIdx.x * 16, nt = blockIdx.y * 16;
  bool isLin = (blockIdx.z != 0);
  const float* W = ws + WS_WPAD + (isLin ? KPAD * HDIM : 0);
  const float* feat = ws + WS_FEAT;
  int mrow = mt + (lane & 15);
  int kb = (lane >> 4) * 2;
  int ncol = nt + (lane & 15);
  const float* arow = feat + mrow * KPAD + kb;   // 8B aligned (kb even)
  const float* bcol = W + kb * HDIM + ncol;
  v8f acc = {};
#pragma unroll 4
  for (int k0 = 0; k0 < KPAD; k0 += 4) {
    v2f a = *(const v2f*)(arow + k0);
    v2f b;
    b.x = bcol[k0 * HDIM];
    b.y = bcol[k0 * HDIM + HDIM];
    acc = __builtin_amdgcn_wmma_f32_16x16x4_f32(
        false, a, false, b, (short)0, acc, false, false);
  }
  float* dst = ws + (isLin ? WS_XL : WS_XW);
  float bias = isLin ? lin1_b[ncol] : 0.f;
  int mbase = mt + ((lane >> 4) << 3);
#pragma unroll
  for (int v = 0; v < 8; ++v)
    dst[(mbase + v) * HDIM + ncol] = acc[v] + bias;
}

__global__ void k_agginit(float* __restrict__ ws) {
  int idx = blockIdx.x * blockDim.x + threadIdx.x;
  if (idx < N_NODES * HDIM) {
    int n = idx >> 7;
    float dis = ws[WS_DIS + n];
    ws[WS_AGG + idx] = ws[WS_XW + idx] * dis * dis;  // self-loop term
  }
}

// Edge scatter-add: one thread handles 4 features of one edge.
__global__ void k_edge(const int* __restrict__ ei, float* __restrict__ ws) {
  int t = blockIdx.x * blockDim.x + threadIdx.x;
  if (t >= NEDGE * 32) return;
  int e = t >> 5;
  int g = (t & 31) << 2;
  int r = ei[e], c = ei[NEDGE + e];
  float nrm = ws[WS_DIS + r] * ws[WS_DIS + c];
  const float4 xv = *(const float4*)(ws + WS_XW + r * HDIM + g);
  float* ab = ws + WS_AGG + c * HDIM + g;
  atomicAdd(ab + 0, xv.x * nrm);
  atomicAdd(ab + 1, xv.y * nrm);
  atomicAdd(ab + 2, xv.z * nrm);
  atomicAdd(ab + 3, xv.w * nrm);
}

__global__ void k_tanh(const float* __restrict__ gcn_b, float* __restrict__ ws) {
  int idx = blockIdx.x * blockDim.x + threadIdx.x;
  if (idx < N_NODES * HDIM)
    ws[WS_Z + idx] = tanhf(ws[WS_AGG + idx] + gcn_b[idx & 127]);
}

// WMMA GEMM: z [N,128] x gate1_w [128,128]; fused sigmoid/gate/relu + bn4 stats.
__global__ void k_gemm_gate(const float* __restrict__ gate_w,
                            const float* __restrict__ gate_b,
                            float* __restrict__ out, float* __restrict__ ws) {
  int lane = threadIdx.x;
  int mt = blockIdx.x * 16, nt = blockIdx.y * 16;
  const float* z = ws + WS_Z;
  int mrow = mt + (lane & 15);
  int kb = (lane >> 4) * 2;
  int ncol = nt + (lane & 15);
  const float* arow = z + mrow * HDIM + kb;      // 8B aligned
  const float* bcol = gate_w + kb * HDIM + ncol;
  v8f acc = {};
#pragma unroll 4
  for (int k0 = 0; k0 < HDIM; k0 += 4) {
    v2f a = *(const v2f*)(arow + k0);
    v2f b;
    b.x = bcol[k0 * HDIM];
    b.y = bcol[k0 * HDIM + HDIM];
    acc = __builtin_amdgcn_wmma_f32_16x16x4_f32(
        false, a, false, b, (short)0, acc, false, false);
  }
  const float* xl = ws + WS_XL;
  float gb = gate_b[ncol];
  int mbase = mt + ((lane >> 4) << 3);
  float s = 0.f, sq = 0.f;
#pragma unroll
  for (int v = 0; v < 8; ++v) {
    int m = mbase + v;
    float zz = z[m * HDIM + ncol];
    float pre = acc[v] + gb;
    float g = 1.f / (1.f + __expf(-pre));
    float o = fmaxf(0.f, (1.f - g) * xl[m * HDIM + ncol] + g * zz);
    out[m * HDIM + ncol] = o;
    s += o;
    sq += o * o;
  }
  // combine the two half-waves (same column set), then 16 atomics per wave
  s += __shfl_down(s, 16, 32);
  sq += __shfl_down(sq, 16, 32);
  if (lane < 16) {
    atomicAdd(&ws[S_BN4SUM + ncol], s);
    atomicAdd(&ws[S_BN4SSQ + ncol], sq);
  }
}

__global__ void k_bn4(const float* __restrict__ g4, const float* __restrict__ b4,
                      float* __restrict__ out, const float* __restrict__ ws) {
  int idx = blockIdx.x * blockDim.x + threadIdx.x;
  if (idx < N_NODES * HDIM) {
    int c = idx & 127;
    float mean = ws[S_BN4SUM + c] / (float)N_NODES;
    float var = ws[S_BN4SSQ + c] / (float)N_NODES - mean * mean;
    out[idx] = (out[idx] - mean) * rsqrtf(var + EPS) * g4[c] + b4[c];
  }
}

// ---------------- launch ----------------
extern "C" void kernel_launch(void* const* d_in, const int* in_sizes, int n_in,
                              void* d_out, int out_size, void* d_ws, size_t ws_size,
                              hipStream_t stream) {
  const float* x    = (const float*)d_in[0];
  const int*   ei   = (const int*)d_in[1];
  const float* w1   = (const float*)d_in[2];
  const float* b1c  = (const float*)d_in[3];
  const float* w2   = (const float*)d_in[4];
  const float* b2c  = (const float*)d_in[5];
  const float* w3   = (const float*)d_in[6];
  const float* b3c  = (const float*)d_in[7];
  const float* g1   = (const float*)d_in[8];
  const float* bb1  = (const float*)d_in[9];
  const float* g2   = (const float*)d_in[10];
  const float* bb2  = (const float*)d_in[11];
  const float* g3   = (const float*)d_in[12];
  const float* bb3  = (const float*)d_in[13];
  const float* gcnw = (const float*)d_in[14];
  const float* gcnb = (const float*)d_in[15];
  const float* linw = (const float*)d_in[16];
  const float* linb = (const float*)d_in[17];
  const float* gatw = (const float*)d_in[18];
  const float* gatb = (const float*)d_in[19];
  const float* g4   = (const float*)d_in[20];
  const float* b4   = (const float*)d_in[21];
  float* out = (float*)d_out;
  float* ws  = (float*)d_ws;

  k_zero<<<2, 256, 0, stream>>>(ws);
  k_padw<<<(2 * KPAD * HDIM + 255) / 256, 256, 0, stream>>>(gcnw, linw, ws);
  k_xstats<<<4096, 256, 0, stream>>>(x, ws);
  k_prep1<<<1, 32, 0, stream>>>(w1, b1c, w2, b2c, g1, bb1, ws);
  k_conv2<<<(N_NODES * L2 + 255) / 256, 256, 0, stream>>>(x, ws);
  k_prep2<<<1, 32, 0, stream>>>(g2, bb2, ws);
  k_conv3<<<(N_NODES * L3 + 255) / 256, 256, 0, stream>>>(w3, b3c, ws);
  k_prep3<<<1, 32, 0, stream>>>(g3, bb3, ws);
  k_featbn<<<(N_NODES * KPAD + 255) / 256, 256, 0, stream>>>(ws);
  k_deginit<<<(N_NODES + 255) / 256, 256, 0, stream>>>(ws);
  k_degacc<<<(NEDGE + 255) / 256, 256, 0, stream>>>(ei, ws);
  k_degfin<<<(N_NODES + 255) / 256, 256, 0, stream>>>(ws);
  k_gemm_feat<<<dim3(625, 8, 2), 32, 0, stream>>>(linb, ws);
  k_agginit<<<(N_NODES * HDIM + 255) / 256, 256, 0, stream>>>(ws);
  k_edge<<<(NEDGE * 32 + 255) / 256, 256, 0, stream>>>(ei, ws);
  k_tanh<<<(N_NODES * HDIM + 255) / 256, 256, 0, stream>>>(gcnb, ws);
  k_gemm_gate<<<dim3(625, 8), 32, 0, stream>>>(gatw, gatb, out, ws);
  k_bn4<<<(N_NODES * HDIM + 255) / 256, 256, 0, stream>>>(g4, b4, out, ws);
}